// HeteroData_GNNmodel_12077448036418
// MI455X (gfx1250) — compile-verified
//
#include <hip/hip_runtime.h>

typedef __attribute__((ext_vector_type(2))) float v2f;
typedef __attribute__((ext_vector_type(8))) float v8f;

// ---------------- problem constants (match reference) ----------------
#define NU 100000
#define NI 50000
#define NE 640000
#define NL 200000
#define F1 128
#define F2 64

// ---------------- workspace layout (float offsets) -------------------
// Layer-1 use               Layer-2 reuse
#define OFF_A   0LL          // t_u  [NU*128]      t2_u [NU*64]
#define OFF_B   12800000LL   // t_i  [NI*128]      t2_i [NI*64]
#define OFF_C   19200000LL   // r_i  [NI*128]      r2_i [NI*64]
#define OFF_D   25600000LL   // r_u  [NU*128]      r2_u [NU*64]
#define OFF_E   38400000LL   // acc_i[NI*128]      acc2_i -> h_i2 [NI*64]
#define OFF_F   44800000LL   // acc_u[NU*128]      acc2_u -> h_u2 [NU*64]
#define OFF_HI1 57600000LL   // h_i1 [NI*128] (persistent across layer 2)
#define OFF_HU1 64000000LL   // h_u1 [NU*128] (persistent across layer 2)
#define OFF_DGI 76800000LL   // deg_i [NI] (int)
#define OFF_DGU 76850000LL   // deg_u [NU] (int)

// ---------------- degree counts --------------------------------------
__global__ void deg_kernel(const int* __restrict__ src, const int* __restrict__ dst,
                           int* __restrict__ deg_u, int* __restrict__ deg_i, int E) {
    int e = blockIdx.x * blockDim.x + threadIdx.x;
    if (e >= E) return;
    atomicAdd(&deg_u[src[e]], 1);
    atomicAdd(&deg_i[dst[e]], 1);
}

// ---------------- WMMA GEMM: C[N,Fout] = X[N,128] @ W[128,Fout] (+bias) ----
// block = 256 threads (8 waves). grid = (ceil(N/128), Fout/16).
// W column tile (128x16 = 8KB) staged in LDS; each wave owns a 16x16 C tile
// and iterates K in steps of 4 with V_WMMA_F32_16X16X4_F32.
__global__ __launch_bounds__(256) void gemm_k128_wmma(
    const float* __restrict__ X, const float* __restrict__ W,
    const float* __restrict__ bias, float* __restrict__ C,
    int N, int Fout)
{
    __shared__ float ldsW[128 * 16];
    const int tid = threadIdx.x;
    const int colBase = blockIdx.y * 16;

    // cooperative stage of W[:, colBase:colBase+16] (Fout is a multiple of 16)
#pragma unroll
    for (int i = 0; i < 8; ++i) {
        int e = tid + i * 256;          // 0..2047
        int k = e >> 4, n = e & 15;
        ldsW[e] = W[k * Fout + colBase + n];
    }
    __syncthreads();

    const int wave = tid >> 5;
    const int lane = tid & 31;
    const int hi   = lane >> 4;         // lane half (0/1)
    const int l16  = lane & 15;
    const int rowBase = blockIdx.x * 128 + wave * 16;
    if (rowBase >= N) return;           // wave-uniform: EXEC all-ones for WMMA

    const float* xrow = X + (size_t)(rowBase + l16) * 128;

    v8f acc = {0.f, 0.f, 0.f, 0.f, 0.f, 0.f, 0.f, 0.f};
#pragma unroll 4
    for (int ks = 0; ks < 32; ++ks) {
        const int k = ks * 4 + hi * 2;
        // A 16x4 fragment: lanes 0-15 hold K=0,1; lanes 16-31 hold K=2,3 (ISA 7.12.2)
        v2f a = *(const v2f*)(xrow + k);
        // B 4x16 fragment: VGPR v holds K=v (lanes 0-15) / K=v+2 (lanes 16-31)
        v2f b;
        b[0] = ldsW[(k + 0) * 16 + l16];
        b[1] = ldsW[(k + 1) * 16 + l16];
        acc = __builtin_amdgcn_wmma_f32_16x16x4_f32(
            false, a, false, b, (short)0, acc, false, false);
    }

    const float bv = bias ? bias[colBase + l16] : 0.0f;
#pragma unroll
    for (int v = 0; v < 8; ++v) {
        const int r = rowBase + v + hi * 8;     // C/D layout: M=v (lo half) / v+8 (hi half)
        C[(size_t)r * Fout + colBase + l16] = acc[v] + bv;
    }
}

// ---------------- edge scatter, 128-dim messages, both directions ----------
__global__ void scatter128(const float* __restrict__ t_u, const float* __restrict__ t_i,
                           const int* __restrict__ src, const int* __restrict__ dst,
                           float* __restrict__ acc_i, float* __restrict__ acc_u, int E)
{
    long long gid = (long long)blockIdx.x * blockDim.x + threadIdx.x;
    if (gid >= (long long)E * 32) return;
    const int e = (int)(gid >> 5);
    const int c = ((int)gid & 31) << 2;
    const int s = src[e], d = dst[e];

    const float4 mu = *(const float4*)(t_u + (size_t)s * 128 + c);
    float* pi = acc_i + (size_t)d * 128 + c;
    atomicAdd(pi + 0, mu.x); atomicAdd(pi + 1, mu.y);
    atomicAdd(pi + 2, mu.z); atomicAdd(pi + 3, mu.w);

    const float4 mi = *(const float4*)(t_i + (size_t)d * 128 + c);
    float* pu = acc_u + (size_t)s * 128 + c;
    atomicAdd(pu + 0, mi.x); atomicAdd(pu + 1, mi.y);
    atomicAdd(pu + 2, mi.z); atomicAdd(pu + 3, mi.w);
}

// ---------------- edge scatter, 64-dim messages (layer 2) ------------------
__global__ void scatter64(const float* __restrict__ t_u, const float* __restrict__ t_i,
                          const int* __restrict__ src, const int* __restrict__ dst,
                          float* __restrict__ acc_i, float* __restrict__ acc_u, int E)
{
    long long gid = (long long)blockIdx.x * blockDim.x + threadIdx.x;
    if (gid >= (long long)E * 16) return;
    const int e = (int)(gid >> 4);
    const int c = ((int)gid & 15) << 2;
    const int s = src[e], d = dst[e];

    const float4 mu = *(const float4*)(t_u + (size_t)s * 64 + c);
    float* pi = acc_i + (size_t)d * 64 + c;
    atomicAdd(pi + 0, mu.x); atomicAdd(pi + 1, mu.y);
    atomicAdd(pi + 2, mu.z); atomicAdd(pi + 3, mu.w);

    const float4 mi = *(const float4*)(t_i + (size_t)d * 64 + c);
    float* pu = acc_u + (size_t)s * 64 + c;
    atomicAdd(pu + 0, mi.x); atomicAdd(pu + 1, mi.y);
    atomicAdd(pu + 2, mi.z); atomicAdd(pu + 3, mi.w);
}

// ---------------- h = (relu?)(acc / max(deg,1) + r) ------------------------
__global__ void combine_kernel(const float* __restrict__ acc, const float* __restrict__ r,
                               const int* __restrict__ deg, float* __restrict__ h,
                               long long total, int fshift, int do_relu)
{
    long long i = (long long)blockIdx.x * blockDim.x + threadIdx.x;
    if (i >= total) return;
    const int node = (int)(i >> fshift);
    const float d = fmaxf((float)deg[node], 1.0f);
    float v = acc[i] / d + r[i];
    h[i] = do_relu ? fmaxf(v, 0.0f) : v;
}

// ---------------- pred[l] = dot64(h_u2[el_src], h_i2[el_dst]) --------------
__global__ void predict_kernel(const float* __restrict__ hu, const float* __restrict__ hi,
                               const int* __restrict__ es, const int* __restrict__ ed,
                               float* __restrict__ out, int L)
{
    int l = blockIdx.x * blockDim.x + threadIdx.x;
    if (l >= L) return;
    const float4* pu = (const float4*)(hu + (size_t)es[l] * 64);
    const float4* pi = (const float4*)(hi + (size_t)ed[l] * 64);
    float s = 0.0f;
#pragma unroll
    for (int j = 0; j < 16; ++j) {
        float4 a = pu[j], b = pi[j];
        s += a.x * b.x + a.y * b.y + a.z * b.z + a.w * b.w;
    }
    out[l] = s;
}

// ---------------------------------------------------------------------------
extern "C" void kernel_launch(void* const* d_in, const int* in_sizes, int n_in,
                              void* d_out, int out_size, void* d_ws, size_t ws_size,
                              hipStream_t stream)
{
    const float* x_user = (const float*)d_in[0];
    const float* x_item = (const float*)d_in[1];
    const int*   src_u  = (const int*)d_in[2];
    const int*   dst_i  = (const int*)d_in[3];
    const int*   el_src = (const int*)d_in[4];
    const int*   el_dst = (const int*)d_in[5];
    const float* Wl_ui1 = (const float*)d_in[6];
    const float* b_ui1  = (const float*)d_in[7];
    const float* Wr_ui1 = (const float*)d_in[8];
    const float* Wl_iu1 = (const float*)d_in[9];
    const float* b_iu1  = (const float*)d_in[10];
    const float* Wr_iu1 = (const float*)d_in[11];
    const float* Wl_ui2 = (const float*)d_in[12];
    const float* b_ui2  = (const float*)d_in[13];
    const float* Wr_ui2 = (const float*)d_in[14];
    const float* Wl_iu2 = (const float*)d_in[15];
    const float* b_iu2  = (const float*)d_in[16];
    const float* Wr_iu2 = (const float*)d_in[17];
    float* out = (float*)d_out;

    float* ws = (float*)d_ws;
    float* t_u   = ws + OFF_A;   // layer2: t2_u
    float* t_i   = ws + OFF_B;   // layer2: t2_i
    float* r_i   = ws + OFF_C;   // layer2: r2_i
    float* r_u   = ws + OFF_D;   // layer2: r2_u
    float* acc_i = ws + OFF_E;   // layer2: acc2_i -> h_i2
    float* acc_u = ws + OFF_F;   // layer2: acc2_u -> h_u2
    float* h_i1  = ws + OFF_HI1;
    float* h_u1  = ws + OFF_HU1;
    int*   deg_i = (int*)(ws + OFF_DGI);
    int*   deg_u = (int*)(ws + OFF_DGU);

    const dim3 blk(256);

    // ---- degrees + zero accumulators (must re-zero every call) ----
    hipMemsetAsync(acc_i, 0, (size_t)NI * F1 * sizeof(float), stream);
    hipMemsetAsync(acc_u, 0, (size_t)NU * F1 * sizeof(float), stream);
    hipMemsetAsync(deg_i, 0, (size_t)NI * sizeof(int), stream);
    hipMemsetAsync(deg_u, 0, (size_t)NU * sizeof(int), stream);
    deg_kernel<<<(NE + 255) / 256, blk, 0, stream>>>(src_u, dst_i, deg_u, deg_i, NE);

    // ---- layer 1: transform-then-aggregate (mean is linear) ----
    dim3 gU1((NU + 127) / 128, F1 / 16), gI1((NI + 127) / 128, F1 / 16);
    gemm_k128_wmma<<<gU1, blk, 0, stream>>>(x_user, Wl_ui1, nullptr, t_u, NU, F1);
    gemm_k128_wmma<<<gI1, blk, 0, stream>>>(x_item, Wl_iu1, nullptr, t_i, NI, F1);
    gemm_k128_wmma<<<gI1, blk, 0, stream>>>(x_item, Wr_ui1, b_ui1,   r_i, NI, F1);
    gemm_k128_wmma<<<gU1, blk, 0, stream>>>(x_user, Wr_iu1, b_iu1,   r_u, NU, F1);

    {
        long long tot = (long long)NE * 32;
        scatter128<<<(unsigned)((tot + 255) / 256), blk, 0, stream>>>(
            t_u, t_i, src_u, dst_i, acc_i, acc_u, NE);
    }
    combine_kernel<<<(unsigned)(((long long)NI * F1 + 255) / 256), blk, 0, stream>>>(
        acc_i, r_i, deg_i, h_i1, (long long)NI * F1, 7, 1);
    combine_kernel<<<(unsigned)(((long long)NU * F1 + 255) / 256), blk, 0, stream>>>(
        acc_u, r_u, deg_u, h_u1, (long long)NU * F1, 7, 1);

    // ---- layer 2 (64-dim messages: half the edge traffic) ----
    hipMemsetAsync(acc_i, 0, (size_t)NI * F2 * sizeof(float), stream);
    hipMemsetAsync(acc_u, 0, (size_t)NU * F2 * sizeof(float), stream);

    dim3 gU2((NU + 127) / 128, F2 / 16), gI2((NI + 127) / 128, F2 / 16);
    gemm_k128_wmma<<<gU2, blk, 0, stream>>>(h_u1, Wl_ui2, nullptr, t_u, NU, F2);
    gemm_k128_wmma<<<gI2, blk, 0, stream>>>(h_i1, Wl_iu2, nullptr, t_i, NI, F2);
    gemm_k128_wmma<<<gI2, blk, 0, stream>>>(h_i1, Wr_ui2, b_ui2,   r_i, NI, F2);
    gemm_k128_wmma<<<gU2, blk, 0, stream>>>(h_u1, Wr_iu2, b_iu2,   r_u, NU, F2);

    {
        long long tot = (long long)NE * 16;
        scatter64<<<(unsigned)((tot + 255) / 256), blk, 0, stream>>>(
            t_u, t_i, src_u, dst_i, acc_i, acc_u, NE);
    }
    // combine in-place: h_i2 lives in acc_i, h_u2 lives in acc_u
    combine_kernel<<<(unsigned)(((long long)NI * F2 + 255) / 256), blk, 0, stream>>>(
        acc_i, r_i, deg_i, acc_i, (long long)NI * F2, 6, 0);
    combine_kernel<<<(unsigned)(((long long)NU * F2 + 255) / 256), blk, 0, stream>>>(
        acc_u, r_u, deg_u, acc_u, (long long)NU * F2, 6, 0);

    // ---- link prediction ----
    predict_kernel<<<(NL + 255) / 256, blk, 0, stream>>>(
        acc_u, acc_i, el_src, el_dst, out, NL);
}